// HIPABlock_4045859193489
// MI455X (gfx1250) — compile-verified
//
#include <hip/hip_runtime.h>
#include <stdint.h>

#define B_    16
#define C_    256
#define H_    160
#define W_    160
#define N_    (H_ * W_)        // 25600
#define COUT_ 256
#define K_    7680             // min(max(8, int(25600*0.3)), 25600)
#define MSORT 32768            // next pow2 >= N_
#define LN_EPS 1e-5f

typedef float v2f __attribute__((ext_vector_type(2)));
typedef float v8f __attribute__((ext_vector_type(8)));

// ---------------------------------------------------------------- zero out map
__global__ void zero_out_kernel(float4* __restrict__ p, size_t n4) {
    size_t i = (size_t)blockIdx.x * blockDim.x + threadIdx.x;
    size_t stride = (size_t)gridDim.x * blockDim.x;
    float4 z = make_float4(0.f, 0.f, 0.f, 0.f);
    for (; i < n4; i += stride) p[i] = z;
}

// ---------------------------------------------------------------- importance
__global__ void importance_kernel(const float* __restrict__ x, float* __restrict__ imp) {
    int i = blockIdx.x * blockDim.x + threadIdx.x;   // over B*N
    if (i >= B_ * N_) return;
    int b = i / N_;
    int n = i - b * N_;
    const float* px = x + (size_t)b * C_ * N_ + n;
    float s = 0.f;
#pragma unroll 8
    for (int c = 0; c < C_; ++c) {
        float v = px[(size_t)c * N_];
        s += v * v;
    }
    imp[i] = fmaxf(sqrtf(s), 1e-8f);
}

// ---------------------------------------------------------------- exact top-K
// One workgroup per batch. Bitonic sort of 32768 64-bit keys in 256 KB LDS
// (CDNA5: up to 320 KB per workgroup). key = bits(imp)<<32 | ~index, sorted
// descending -> value desc, index asc on ties (matches jax.lax.top_k).
__global__ __launch_bounds__(1024) void topk_sort_kernel(const float* __restrict__ imp,
                                                         int* __restrict__ topk) {
    extern __shared__ unsigned long long keys[];   // MSORT entries
    const int b = blockIdx.x;
    const int t = threadIdx.x;

    for (int i = t; i < MSORT; i += 1024) {
        unsigned long long key = 0ull;             // pad sorts last (imp >= 1e-8)
        if (i < N_) {
            unsigned int fb = __float_as_uint(imp[b * N_ + i]);  // positive floats: bit order == value order
            key = ((unsigned long long)fb << 32) | (unsigned int)(~(unsigned int)i);
        }
        keys[i] = key;
    }
    __syncthreads();

    for (int k = 2; k <= MSORT; k <<= 1) {
        for (int j = k >> 1; j > 0; j >>= 1) {
            for (int i = t; i < MSORT; i += 1024) {
                int ij = i ^ j;
                if (ij > i) {
                    unsigned long long a = keys[i];
                    unsigned long long c = keys[ij];
                    bool up = ((i & k) == 0);      // 'up' regions sorted descending
                    if ((a < c) == up) { keys[i] = c; keys[ij] = a; }
                }
            }
            __syncthreads();
        }
    }

    for (int kk = t; kk < K_; kk += 1024) {
        unsigned int low = (unsigned int)(keys[kk] & 0xFFFFFFFFull);
        topk[b * K_ + kk] = (int)(~low);
    }
}

// ---------------------------------------------------------------- coords + sparsity
__global__ void coords_kernel(const int* __restrict__ topk, float* __restrict__ coords,
                              float* __restrict__ sparsity) {
    int i = blockIdx.x * blockDim.x + threadIdx.x;   // over B*K
    if (i < B_ * K_) {
        int idx = topk[i];
        float xs = (float)(idx % W_) / (float)W_;
        float ys = (float)(idx / W_) / (float)H_;
        float4 cc = make_float4(xs, ys, 1.f / (float)W_, 1.f / (float)H_);
        ((float4*)coords)[i] = cc;
    }
    if (i == 0) *sparsity = (float)K_ / (float)N_;
}

// ---------------------------------------------------------------- gather + LN + WMMA GEMM + scatter
// Block: 256 threads (8 waves of 32), one 16-row tile of kept tokens.
// Each wave owns two 16-col output tiles (8 waves * 32 cols = 256 = Cout).
#define APAD 260   // row pad: stride 260 dwords -> bank stride 4, conflict-free; keeps 8B align
__global__ __launch_bounds__(256) void ln_gemm_scatter_kernel(
    const float* __restrict__ x, const int* __restrict__ topk,
    const float* __restrict__ gamma, const float* __restrict__ beta,
    const float* __restrict__ wproj, const float* __restrict__ bproj,
    float* __restrict__ out, float* __restrict__ kept_feat) {
    __shared__ float A[16][APAD];   // normalized kept rows (in-place LN)
    __shared__ int   idxs[16];

    const int b     = blockIdx.y;
    const int tileM = blockIdx.x;        // 0 .. K_/16-1
    const int t     = threadIdx.x;
    const int k0    = tileM * 16;

    if (t < 16) idxs[t] = topk[b * K_ + k0 + t];
    __syncthreads();

    // gather kept[m][c] = x[b, c, idx[m]]
    const float* xb = x + (size_t)b * C_ * N_;
    for (int i = t; i < 16 * C_; i += 256) {
        int m = i & 15;
        int c = i >> 4;
        A[m][c] = xb[(size_t)c * N_ + idxs[m]];
    }
    __syncthreads();

    const int wave = t >> 5;
    const int lane = t & 31;

    // LayerNorm: wave w normalizes rows 2w, 2w+1 (wave32 shuffle reduction)
    for (int r = 0; r < 2; ++r) {
        int m = wave * 2 + r;
        float s1 = 0.f, s2 = 0.f;
        for (int c = lane; c < C_; c += 32) {
            float v = A[m][c];
            s1 += v;
            s2 += v * v;
        }
        for (int off = 16; off > 0; off >>= 1) {
            s1 += __shfl_xor(s1, off, 32);
            s2 += __shfl_xor(s2, off, 32);
        }
        float mu  = s1 * (1.f / C_);
        float var = s2 * (1.f / C_) - mu * mu;    // biased variance
        float rs  = rsqrtf(var + LN_EPS);
        for (int c = lane; c < C_; c += 32)
            A[m][c] = (A[m][c] - mu) * rs * gamma[c] + beta[c];
    }
    __syncthreads();

    // WMMA f32 16x16x4: D[16x16] += A[16x4] x B[4x16], K-loop over C in steps of 4.
    // Lane layout (ISA 7.12.2): lanes 0-15 carry K={kk,kk+1}, lanes 16-31 carry K={kk+2,kk+3}.
    const int half = lane >> 4;          // 0 or 1
    const int l16  = lane & 15;
    const int o0   = wave * 32;
    const int o1   = o0 + 16;

    v8f acc0 = {};
    v8f acc1 = {};
    const float* wp0 = wproj + (size_t)(o0 + l16) * C_ + half * 2;  // B[k][o] = wproj[o*C + k]
    const float* wp1 = wproj + (size_t)(o1 + l16) * C_ + half * 2;

    for (int kk = 0; kk < C_; kk += 4) {
        int ka = kk + half * 2;
        v2f a  = *(const v2f*)&A[l16][ka];
        v2f b0 = *(const v2f*)(wp0 + kk);
        v2f b1 = *(const v2f*)(wp1 + kk);
        acc0 = __builtin_amdgcn_wmma_f32_16x16x4_f32(false, a, false, b0, (short)0, acc0, false, false);
        acc1 = __builtin_amdgcn_wmma_f32_16x16x4_f32(false, a, false, b1, (short)0, acc1, false, false);
    }

    // Epilogue: +bias, write kept_feat [B,K,Cout] and scatter into out [B,Cout,N].
    // C/D layout: lane<16 -> M=v, lane>=16 -> M=8+v, N = lane&15.
    float bp0 = bproj[o0 + l16];
    float bp1 = bproj[o1 + l16];
    float* kf = kept_feat + ((size_t)b * K_ + k0) * COUT_;
    float* ob = out + (size_t)b * COUT_ * N_;
#pragma unroll
    for (int v = 0; v < 8; ++v) {
        int m = half * 8 + v;
        float r0 = acc0[v] + bp0;
        float r1 = acc1[v] + bp1;
        kf[(size_t)m * COUT_ + o0 + l16] = r0;
        kf[(size_t)m * COUT_ + o1 + l16] = r1;
        int n = idxs[m];
        ob[(size_t)(o0 + l16) * N_ + n] = r0;
        ob[(size_t)(o1 + l16) * N_ + n] = r1;
    }
}

// ---------------------------------------------------------------- launch
extern "C" void kernel_launch(void* const* d_in, const int* in_sizes, int n_in,
                              void* d_out, int out_size, void* d_ws, size_t ws_size,
                              hipStream_t stream) {
    const float* x     = (const float*)d_in[0];
    const float* gamma = (const float*)d_in[1];
    const float* beta  = (const float*)d_in[2];
    const float* wproj = (const float*)d_in[3];
    const float* bproj = (const float*)d_in[4];

    // d_out layout: out | kept_feat | coords | sparsity (flat, return order)
    const size_t OUT_ELEMS   = (size_t)B_ * COUT_ * N_;          // 104,857,600
    const size_t KEPT_ELEMS  = (size_t)B_ * K_ * COUT_;          //  31,457,280
    const size_t COORD_ELEMS = (size_t)B_ * K_ * 4;              //     491,520
    float* out_map   = (float*)d_out;
    float* kept_feat = out_map + OUT_ELEMS;
    float* coords    = kept_feat + KEPT_ELEMS;
    float* sparsity  = coords + COORD_ELEMS;

    // workspace: imp [B*N] f32, topk [B*K] i32
    float* imp  = (float*)d_ws;
    int*   topk = (int*)((char*)d_ws + (size_t)B_ * N_ * sizeof(float));

    // 1) zero the dense output map (harness poisons once; re-zero every call)
    zero_out_kernel<<<4096, 256, 0, stream>>>((float4*)out_map, OUT_ELEMS / 4);

    // 2) per-pixel L2 importance
    importance_kernel<<<(B_ * N_ + 255) / 256, 256, 0, stream>>>(x, imp);

    // 3) exact per-batch top-K (bitonic sort, 256 KB dynamic LDS per workgroup)
    const size_t sort_lds = (size_t)MSORT * sizeof(unsigned long long);   // 262144 B
    hipFuncSetAttribute((const void*)topk_sort_kernel,
                        hipFuncAttributeMaxDynamicSharedMemorySize, (int)sort_lds);
    topk_sort_kernel<<<B_, 1024, sort_lds, stream>>>(imp, topk);

    // 4) coords + sparsity
    coords_kernel<<<(B_ * K_ + 255) / 256, 256, 0, stream>>>(topk, coords, sparsity);

    // 5) gather + LayerNorm + WMMA projection + scatter
    dim3 grid(K_ / 16, B_);
    ln_gemm_scatter_kernel<<<grid, 256, 0, stream>>>(x, topk, gamma, beta, wproj, bproj,
                                                     out_map, kept_feat);
}